// SEBlock_34445637714659
// MI455X (gfx1250) — compile-verified
//
#include <hip/hip_runtime.h>
#include <math.h>

typedef __attribute__((ext_vector_type(2))) float v2f;
typedef __attribute__((ext_vector_type(8))) float v8f;

#define BATCH    8
#define CH       256
#define HW       (192 * 192)   // 36864 elements per plane
#define HW4      (HW / 4)      // 9216 float4 per plane
#define NPLANE   (BATCH * CH)  // 2048
#define INTERNAL 16

// ---------------------------------------------------------------------------
// Kernel 1: sumsq[b*C+c] = sum over HxW of x^2   (one block per plane)
// ---------------------------------------------------------------------------
__global__ __launch_bounds__(256) void k_sumsq(const float* __restrict__ in,
                                               float* __restrict__ ssq) {
  const int plane = blockIdx.x;
  const float4* p = (const float4*)in + (size_t)plane * HW4;
  float acc = 0.f;
  for (int i = threadIdx.x; i < HW4; i += 256) {
    float4 v = p[i];
    acc = fmaf(v.x, v.x, acc);
    acc = fmaf(v.y, v.y, acc);
    acc = fmaf(v.z, v.z, acc);
    acc = fmaf(v.w, v.w, acc);
  }
  // wave32 reduce
  #pragma unroll
  for (int off = 16; off > 0; off >>= 1) acc += __shfl_xor(acc, off, 32);
  __shared__ float red[8];
  if ((threadIdx.x & 31) == 0) red[threadIdx.x >> 5] = acc;
  __syncthreads();
  if (threadIdx.x == 0) {
    float s = 0.f;
    #pragma unroll
    for (int w = 0; w < 8; ++w) s += red[w];
    ssq[plane] = s;
  }
}

// ---------------------------------------------------------------------------
// Kernel 2: excitation MLP via V_WMMA_F32_16X16X4_F32 (single block, 16 waves)
//   x[b,c] = Gx/(Gx+1e-6), Gx = sqrt(sumsq)
//   h    = relu(x @ w_down^T + b_down)        [16(pad) x 16]
//   gate = sigmoid(h @ w_up^T + b_up)         [8 x 256]
// ---------------------------------------------------------------------------
__global__ __launch_bounds__(512) void k_gate(const float* __restrict__ ssq,
                                              const float* __restrict__ w_down,
                                              const float* __restrict__ b_down,
                                              const float* __restrict__ w_up,
                                              const float* __restrict__ b_up,
                                              float* __restrict__ gate) {
  __shared__ float x_lds[16 * 256];   // A1: rows 8..15 zero-padded
  __shared__ float wdn[16 * 256];     // w_down [16,256] row-major; B1[k][n] = wdn[n*256+k]
  __shared__ float wup[256 * 16];     // w_up   [256,16] row-major; B2[k][n] = wup[n*16+k]
  __shared__ float h_lds[16 * 16];
  __shared__ float bdn[16];
  __shared__ float bup[256];

  const int tid = threadIdx.x;
  for (int i = tid; i < 16 * 256; i += 512) {
    const int bb = i >> 8;
    float v = 0.f;
    if (bb < BATCH) {
      const float g = sqrtf(ssq[i]);
      v = g / (g + 1e-6f);
    }
    x_lds[i] = v;
    wdn[i]   = w_down[i];
    wup[i]   = w_up[i];
  }
  if (tid < 16)  bdn[tid] = b_down[tid];
  if (tid < 256) bup[tid] = b_up[tid];
  __syncthreads();

  const int lane = tid & 31;
  const int wave = tid >> 5;
  const int hsel = lane >> 4;   // 0: lanes 0-15, 1: lanes 16-31
  const int lr   = lane & 15;   // M index for A, N index for B/C/D

  // ---- matmul 1: [16 x 256] @ [256 x 16], wave 0 only (uniform branch) ----
  if (wave == 0) {
    v8f c = {};
    for (int k0 = 0; k0 < 256; k0 += 4) {
      const int k = k0 + 2 * hsel;      // VGPR0 holds K = k0 + 2*half
      v2f a, b;
      a[0] = x_lds[lr * 256 + k];
      a[1] = x_lds[lr * 256 + k + 1];
      b[0] = wdn[lr * 256 + k];          // B[k][lr] = w_down[lr][k]
      b[1] = wdn[lr * 256 + k + 1];
      c = __builtin_amdgcn_wmma_f32_16x16x4_f32(
          /*neg_a=*/false, a, /*neg_b=*/false, b,
          /*c_mod=*/(short)0, c, /*reuse_a=*/false, /*reuse_b=*/false);
    }
    #pragma unroll
    for (int r = 0; r < 8; ++r) {
      const int m = 8 * hsel + r;        // C/D layout: VGPR r -> M = 8*half + r
      const float h = c[r] + bdn[lr];
      h_lds[m * 16 + lr] = h > 0.f ? h : 0.f;
    }
  }
  __syncthreads();

  // ---- matmul 2: [16 x 16] @ [16 x 16] per N-tile; wave w -> cols 16w.. ----
  {
    const int n0 = wave * 16;
    v8f c = {};
    #pragma unroll
    for (int k0 = 0; k0 < 16; k0 += 4) {
      const int k = k0 + 2 * hsel;
      v2f a, b;
      a[0] = h_lds[lr * 16 + k];
      a[1] = h_lds[lr * 16 + k + 1];
      b[0] = wup[(n0 + lr) * 16 + k];    // B[k][n] = w_up[n][k]
      b[1] = wup[(n0 + lr) * 16 + k + 1];
      c = __builtin_amdgcn_wmma_f32_16x16x4_f32(
          false, a, false, b, (short)0, c, false, false);
    }
    if (hsel == 0) {                      // rows 0..7 are the real batch
      #pragma unroll
      for (int r = 0; r < 8; ++r) {
        const float z = c[r] + bup[n0 + lr];
        gate[r * 256 + n0 + lr] = 1.f / (1.f + __expf(-z));
      }
    }
  }
}

// ---------------------------------------------------------------------------
// Kernel 3: out = in * gate[plane]   (streaming, float4, gate uniform/block)
// ---------------------------------------------------------------------------
__global__ __launch_bounds__(256) void k_scale(const float* __restrict__ in,
                                               const float* __restrict__ gate,
                                               float* __restrict__ out) {
  const int plane = blockIdx.y;
  const float g = gate[plane];
  const size_t idx = (size_t)plane * HW4 + (size_t)blockIdx.x * 256 + threadIdx.x;
  float4 v = ((const float4*)in)[idx];
  v.x *= g; v.y *= g; v.z *= g; v.w *= g;
  ((float4*)out)[idx] = v;
}

// ---------------------------------------------------------------------------
extern "C" void kernel_launch(void* const* d_in, const int* in_sizes, int n_in,
                              void* d_out, int out_size, void* d_ws, size_t ws_size,
                              hipStream_t stream) {
  const float* in  = (const float*)d_in[0];   // [8,256,192,192]
  const float* wdn = (const float*)d_in[1];   // [16,256]
  const float* bdn = (const float*)d_in[2];   // [16]
  const float* wup = (const float*)d_in[3];   // [256,16]
  const float* bup = (const float*)d_in[4];   // [256]
  float* out = (float*)d_out;

  float* ssq  = (float*)d_ws;                 // 2048 floats
  float* gate = ssq + NPLANE;                 // 2048 floats (8*256 used)

  k_sumsq<<<NPLANE, 256, 0, stream>>>(in, ssq);
  k_gate<<<1, 512, 0, stream>>>(ssq, wdn, bdn, wup, bup, gate);
  k_scale<<<dim3(HW4 / 256, NPLANE), 256, 0, stream>>>(in, gate, out);
}